// GraphProjection_3D_8203387535723
// MI455X (gfx1250) — compile-verified
//
#include <hip/hip_runtime.h>
#include <cstdint>
#include <cstddef>

// ---------------------------------------------------------------------------
// GraphProjection_3D for MI455X (gfx1250).
// Memory-bound gather workload: no GEMM structure -> no WMMA (each point has
// distinct gather corners, so there is no shared B operand). CDNA5 features
// used instead: async global->LDS staging of packed occupancy bitmaps
// (ASYNCcnt path), wave32-shaped channel loops, b128 vector gathers.
// ---------------------------------------------------------------------------

constexpr int kGrid         = 64;
constexpr int kWordsPerGrid = kGrid * kGrid * kGrid / 32;  // 8192 u32 = 32 KB
constexpr int kOutW         = 963;                         // 3+64+128+256+500+4+4+4
constexpr int kTPB          = 256;                         // 8 waves (wave32)
constexpr int kPPB          = 128;                         // points per block

#if defined(__AMDGCN__) && __has_builtin(__builtin_amdgcn_global_load_async_to_lds_b128)
#define HAVE_ASYNC_LDS 1
#else
#define HAVE_ASYNC_LDS 0
#endif

#if HAVE_ASYNC_LDS
typedef int v4i __attribute__((ext_vector_type(4)));
typedef __attribute__((address_space(1))) v4i gv4i;  // global
typedef __attribute__((address_space(3))) v4i lv4i;  // LDS
#endif

// ---------------------------------------------------------------------------
__global__ void zero_bits_kernel(unsigned* __restrict__ bits, int n) {
  int i = blockIdx.x * blockDim.x + threadIdx.x;
  if (i < n) bits[i] = 0u;
}

// Scatter occupancy as packed bits (atomicOr on u32); binarization is implicit.
__global__ void voxelize_kernel(const float* __restrict__ pc0,
                                const float* __restrict__ pc1,
                                const float* __restrict__ pc2,
                                unsigned* __restrict__ bits, int M) {
  int i = blockIdx.x * blockDim.x + threadIdx.x;
  if (i >= M) return;
  int g = blockIdx.y;  // cloud index 0..2 (uniform per block)
  const float* pc = (g == 0) ? pc0 : ((g == 1) ? pc1 : pc2);
  float x = pc[3 * i + 0], y = pc[3 * i + 1], z = pc[3 * i + 2];
  float h = fminf(fmaxf(32.0f * y + 32.0f, 0.0f), 63.0f);
  float w = fminf(fmaxf(32.0f * x + 32.0f, 0.0f), 63.0f);
  float c = fminf(fmaxf(32.0f * z + 32.0f, 0.0f), 63.0f);
  int hi = (int)floorf(h), wi = (int)floorf(w), ci = (int)floorf(c);
  int v = (hi * kGrid + wi) * kGrid + ci;
  atomicOr(&bits[g * kWordsPerGrid + (v >> 5)], 1u << (v & 31));
}

// ---------------------------------------------------------------------------
__device__ __forceinline__ int iclamp(int v, int hi) {
  return v < 0 ? 0 : (v > hi ? hi : v);
}

// Bilinear-in-(x,y), ceil-in-z projection over one feature level.
// Feature row strides are multiples of 16B for all levels (C=64/128/256/500),
// so float4 (global_load_b128) gathers are aligned. Output stores are scalar
// (row stride 963 floats is only 4B-aligned).
template <int D, int C4>
__device__ __forceinline__ void proj_level(const float* __restrict__ feat,
                                           float x, float y, float z,
                                           int lane, float* __restrict__ outp) {
  float x1f = floorf(x), x2f = ceilf(x);
  float y1f = floorf(y), y2f = ceilf(y);
  float z1f = ceilf(z);
  int ix1 = iclamp((int)x1f, D - 1);
  int ix2 = iclamp((int)x2f, D - 1);
  int iy1 = iclamp((int)y1f, D - 1);
  int iy2 = iclamp((int)y2f, D - 1);
  int iz1 = iclamp((int)z1f, D - 1);
  float wx2 = x2f - x, wx1 = x - x1f;
  float wy2 = y2f - y, wy1 = y - y1f;
  float w11 = wx2 * wy2, w21 = wx1 * wy2, w12 = wx2 * wy1, w22 = wx1 * wy1;
  constexpr int C = 4 * C4;
  const float4* r11 = (const float4*)(feat + (size_t)((ix1 * D + iy1) * D + iz1) * C);
  const float4* r21 = (const float4*)(feat + (size_t)((ix2 * D + iy1) * D + iz1) * C);
  const float4* r12 = (const float4*)(feat + (size_t)((ix1 * D + iy2) * D + iz1) * C);
  const float4* r22 = (const float4*)(feat + (size_t)((ix2 * D + iy2) * D + iz1) * C);
  for (int c4 = lane; c4 < C4; c4 += 32) {
    float4 a = r11[c4], b = r21[c4], e = r12[c4], d = r22[c4];
    float* po = outp + 4 * c4;
    po[0] = w11 * a.x + w21 * b.x + w12 * e.x + w22 * d.x;
    po[1] = w11 * a.y + w21 * b.y + w12 * e.y + w22 * d.y;
    po[2] = w11 * a.z + w21 * b.z + w12 * e.z + w22 * d.z;
    po[3] = w11 * a.w + w21 * b.w + w12 * e.w + w22 * d.w;
  }
}

// ---------------------------------------------------------------------------
__global__ void __launch_bounds__(kTPB)
project_kernel(const float* __restrict__ coord,
               const float* __restrict__ f0, const float* __restrict__ f1,
               const float* __restrict__ f2, const float* __restrict__ f3,
               const unsigned* __restrict__ bits,
               float* __restrict__ out, int N) {
  __shared__ unsigned sbits[3 * kWordsPerGrid];  // 96 KB of 320 KB WGP LDS

  // ---- stage the three packed occupancy grids into LDS --------------------
  {
    const int nv = 3 * kWordsPerGrid / 4;  // 128-bit transfers
#if HAVE_ASYNC_LDS
    gv4i* g = (gv4i*)bits;     // C-style cast: retype + drop const, AS(1)
    lv4i* s = (lv4i*)sbits;    // AS(3)
    for (int k = threadIdx.x; k < nv; k += kTPB) {
      __builtin_amdgcn_global_load_async_to_lds_b128(g + k, s + k, 0, 0);
    }
#if __has_builtin(__builtin_amdgcn_s_wait_asynccnt)
    __builtin_amdgcn_s_wait_asynccnt(0);
#else
    asm volatile("s_wait_asynccnt 0" ::: "memory");
#endif
#else
    const uint4* g4 = (const uint4*)bits;
    uint4* s4 = (uint4*)sbits;
    for (int k = threadIdx.x; k < nv; k += kTPB) s4[k] = g4[k];
#endif
  }
  __syncthreads();

  const int lane = threadIdx.x & 31;
  const int p0 = blockIdx.x * kPPB;
  int pend = p0 + kPPB;
  if (pend > N) pend = N;

  // one point per wave per iteration; lanes stride over channel groups
  for (int p = p0 + (threadIdx.x >> 5); p < pend; p += kTPB / 32) {
    float X = coord[3 * p + 0], Y = coord[3 * p + 1], Z = coord[3 * p + 2];
    float h = fminf(fmaxf(32.0f * Y + 32.0f, 0.0f), 63.0f);
    float w = fminf(fmaxf(32.0f * X + 32.0f, 0.0f), 63.0f);
    float c = fminf(fmaxf(32.0f * Z + 32.0f, 0.0f), 63.0f);
    float* orow = out + (size_t)p * kOutW;
    if (lane == 0) { orow[0] = X; orow[1] = Y; orow[2] = Z; }

    proj_level<64, 16>(f0, h, w, c, lane, orow + 3);                                   // out1: 64ch
    proj_level<32, 32>(f1, 0.5f * h, 0.5f * w, 0.5f * c, lane, orow + 67);             // out2: 128ch
    proj_level<16, 64>(f2, 0.25f * h, 0.25f * w, 0.25f * c, lane, orow + 195);         // out3: 256ch
    proj_level<8, 125>(f3, 0.125f * h, 0.125f * w, 0.125f * c, lane, orow + 451);      // out4: 500ch

    // ---- out5/6/7: occupancy-grid projection, 4 replicated channels each --
    {
      float x1f = floorf(h), x2f = ceilf(h);
      float y1f = floorf(w), y2f = ceilf(w);
      float z1f = ceilf(c);
      int ix1 = iclamp((int)x1f, 63), ix2 = iclamp((int)x2f, 63);
      int iy1 = iclamp((int)y1f, 63), iy2 = iclamp((int)y2f, 63);
      int iz1 = iclamp((int)z1f, 63);
      float wx2 = x2f - h, wx1 = h - x1f;
      float wy2 = y2f - w, wy1 = w - y1f;
      float w11 = wx2 * wy2, w21 = wx1 * wy2, w12 = wx2 * wy1, w22 = wx1 * wy1;
      if (lane < 12) {
        const unsigned* sb = sbits + (lane >> 2) * kWordsPerGrid;  // grid = lane/4
        int v11 = (ix1 * 64 + iy1) * 64 + iz1;
        int v21 = (ix2 * 64 + iy1) * 64 + iz1;
        int v12 = (ix1 * 64 + iy2) * 64 + iz1;
        int v22 = (ix2 * 64 + iy2) * 64 + iz1;
        float q11 = (float)((sb[v11 >> 5] >> (v11 & 31)) & 1u);
        float q21 = (float)((sb[v21 >> 5] >> (v21 & 31)) & 1u);
        float q12 = (float)((sb[v12 >> 5] >> (v12 & 31)) & 1u);
        float q22 = (float)((sb[v22 >> 5] >> (v22 & 31)) & 1u);
        orow[951 + lane] = w11 * q11 + w21 * q21 + w12 * q12 + w22 * q22;
      }
    }
  }
}

// ---------------------------------------------------------------------------
extern "C" void kernel_launch(void* const* d_in, const int* in_sizes, int n_in,
                              void* d_out, int out_size, void* d_ws, size_t ws_size,
                              hipStream_t stream) {
  const float* coord = (const float*)d_in[0];
  const float* f0    = (const float*)d_in[1];
  const float* f1    = (const float*)d_in[2];
  const float* f2    = (const float*)d_in[3];
  const float* f3    = (const float*)d_in[4];
  const float* pc0   = (const float*)d_in[5];
  const float* pc1   = (const float*)d_in[6];
  const float* pc2   = (const float*)d_in[7];
  float* out = (float*)d_out;
  unsigned* bits = (unsigned*)d_ws;  // 3 * 32 KB packed occupancy bitmaps

  int N = in_sizes[0] / 3;
  int M = in_sizes[5] / 3;

  int nbits = 3 * kWordsPerGrid;
  hipLaunchKernelGGL(zero_bits_kernel, dim3((nbits + 255) / 256), dim3(256), 0,
                     stream, bits, nbits);
  hipLaunchKernelGGL(voxelize_kernel, dim3((M + 255) / 256, 3), dim3(256), 0,
                     stream, pc0, pc1, pc2, bits, M);
  hipLaunchKernelGGL(project_kernel, dim3((N + kPPB - 1) / kPPB), dim3(kTPB), 0,
                     stream, coord, f0, f1, f2, f3, bits, out, N);
}